// MyModel_14242111554120
// MI455X (gfx1250) — compile-verified
//
#include <hip/hip_runtime.h>

// ---------------------------------------------------------------------------
// GNN pipeline for MI455X (gfx1250, wave32):
//   h1 = scatter_add(x[src] -> dst)            (atomic f32, L2-resident)
//   h2 = h1 @ (diag(w1)@Wd1) + b1              (V_WMMA_F32_16X16X4_F32)
//   h3 = scatter_add(h2[src] -> dst)
//   h4 = h3 @ (diag(w2)@Wd2) + b2              (V_WMMA_F32_16X16X4_F32)
//   out = softmax(h4 @ Wout + bout)            (VALU, N=4 too skinny for WMMA)
// ---------------------------------------------------------------------------

typedef __attribute__((ext_vector_type(2))) float v2f;
typedef __attribute__((ext_vector_type(8))) float v8f;

#define N_NODES 50000
#define N_EDGES 600000
#define D_FEAT  128
#define D_H1    64
#define D_H2    32
#define D_OUT   4

// ---------------------------------------------------------------------------
// Fold the per-feature graph scales into the dense weights:
//   Wd1s[k][j] = w1[k] * Wd1[k][j]   (128x64)
//   Wd2s[k][j] = w2[k] * Wd2[k][j]   (64x32)
// ---------------------------------------------------------------------------
__global__ void scale_weights_kernel(const float* __restrict__ w1,
                                     const float* __restrict__ Wd1,
                                     const float* __restrict__ w2,
                                     const float* __restrict__ Wd2,
                                     float* __restrict__ Wd1s,
                                     float* __restrict__ Wd2s) {
  int t = blockIdx.x * blockDim.x + threadIdx.x;
  if (t < D_FEAT * D_H1) {
    int k = t / D_H1;
    Wd1s[t] = w1[k] * Wd1[t];
  }
  int t2 = t - D_FEAT * D_H1;
  if (t2 >= 0 && t2 < D_H1 * D_H2) {
    int k = t2 / D_H2;
    Wd2s[t2] = w2[k] * Wd2[t2];
  }
}

__global__ void zero_kernel(float* __restrict__ p, int n) {
  int i = blockIdx.x * blockDim.x + threadIdx.x;
  int stride = gridDim.x * blockDim.x;
  for (; i < n; i += stride) p[i] = 0.0f;
}

// ---------------------------------------------------------------------------
// SpMM scatter, 128 features: one wave per edge. Each lane gathers a float4
// (coalesced 512B/wave) and atomically accumulates into the dst row.
// ---------------------------------------------------------------------------
__global__ void spmm_scatter128(const float* __restrict__ x,
                                const int* __restrict__ src,
                                const int* __restrict__ dst,
                                float* __restrict__ out) {
  int wave = (blockIdx.x * blockDim.x + threadIdx.x) >> 5;
  int lane = threadIdx.x & 31;
  if (wave >= N_EDGES) return;
  int s = src[wave];
  int d = dst[wave];
  const float4 v = *(const float4*)(x + (size_t)s * D_FEAT + lane * 4);
  float* o = out + (size_t)d * D_FEAT + lane * 4;
  atomicAdd(o + 0, v.x);
  atomicAdd(o + 1, v.y);
  atomicAdd(o + 2, v.z);
  atomicAdd(o + 3, v.w);
}

// SpMM scatter, 64 features: one wave per edge, float2 per lane.
__global__ void spmm_scatter64(const float* __restrict__ h,
                               const int* __restrict__ src,
                               const int* __restrict__ dst,
                               float* __restrict__ out) {
  int wave = (blockIdx.x * blockDim.x + threadIdx.x) >> 5;
  int lane = threadIdx.x & 31;
  if (wave >= N_EDGES) return;
  int s = src[wave];
  int d = dst[wave];
  const float2 v = *(const float2*)(h + (size_t)s * D_H1 + lane * 2);
  float* o = out + (size_t)d * D_H1 + lane * 2;
  atomicAdd(o + 0, v.x);
  atomicAdd(o + 1, v.y);
}

// ---------------------------------------------------------------------------
// FP32 GEMM via V_WMMA_F32_16X16X4_F32.
//   C[M x N] = A[M x K] @ B[K x N] + bias[N]
// One wave per 16x16 output tile; N/16 waves per block share a 16xK A-tile
// staged in LDS (stride K+4 floats -> bank-conflict-free ds_load_b64, and
// 16B-aligned rows for the ds_store_b128 staging).
//
// ISA VGPR layouts (CDNA5 ISA):
//   A 16x4 f32 : lanes 0-15 row M=lane, VGPR{0,1}=K{0,1}; lanes 16-31 K{2,3}
//   B 4x16 f32 : lanes 0-15 col N=lane, VGPR{0,1}=K{0,1}; lanes 16-31 K{2,3}
//   C/D 16x16  : VGPR i, lane half h -> row i+8h, col lane%16
// M must be a multiple of 16 (50000 = 3125*16, exact).
// ---------------------------------------------------------------------------
template <int K, int N>
__global__ void gemm_wmma_f32(const float* __restrict__ A,
                              const float* __restrict__ B,
                              const float* __restrict__ bias,
                              float* __restrict__ C) {
  constexpr int NT  = N / 16;   // waves per block = n-tiles
  constexpr int LDA = K + 4;    // padded LDS row stride (floats)
  __shared__ __align__(16) float As[16 * LDA];

  const int mtile = blockIdx.x;
  const int tid   = threadIdx.x;          // blockDim.x == NT*32
  const int lane  = tid & 31;
  const int half  = lane >> 4;            // 0: K even pair low, 1: high
  const int r     = lane & 15;
  const int wv    = tid >> 5;             // wave id == n-tile index

  // Cooperative stage of the 16xK A-tile into LDS (float4 chunks).
  const float* Ag = A + (size_t)mtile * 16 * K;
  for (int i = tid; i < (16 * K) / 4; i += NT * 32) {
    int row = (i * 4) / K;
    int col = (i * 4) % K;
    float4 v = *(const float4*)(Ag + row * K + col);
    *(float4*)(&As[row * LDA + col]) = v;
  }
  __syncthreads();

  const float* Bn = B + wv * 16 + r;      // lane r -> column (coalesced dwords)
  v8f acc = {};
#pragma unroll
  for (int k0 = 0; k0 < K; k0 += 4) {
    v2f a = *(const v2f*)(&As[r * LDA + k0 + 2 * half]);
    v2f b;
    b.x = Bn[(size_t)(k0 + 2 * half + 0) * N];
    b.y = Bn[(size_t)(k0 + 2 * half + 1) * N];
    acc = __builtin_amdgcn_wmma_f32_16x16x4_f32(
        /*neg_a=*/false, a, /*neg_b=*/false, b,
        /*c_mod=*/(short)0, acc, /*reuse_a=*/false, /*reuse_b=*/false);
  }

  // Epilogue: add column bias, write the 16x16 tile.
  const float bcol = bias[wv * 16 + r];
  float* Cg = C + ((size_t)mtile * 16 + half * 8) * N + wv * 16 + r;
#pragma unroll
  for (int i = 0; i < 8; ++i) {
    Cg[(size_t)i * N] = acc[i] + bcol;
  }
}

// ---------------------------------------------------------------------------
// Final skinny layer (K=32 -> N=4) + numerically-stable softmax, one thread
// per node. Wout/bout are tiny and L1-resident.
// ---------------------------------------------------------------------------
__global__ void out_softmax_kernel(const float* __restrict__ h4,
                                   const float* __restrict__ Wout,
                                   const float* __restrict__ bout,
                                   float* __restrict__ out) {
  int node = blockIdx.x * blockDim.x + threadIdx.x;
  if (node >= N_NODES) return;
  const float* hr = h4 + (size_t)node * D_H2;
  float acc[D_OUT] = {bout[0], bout[1], bout[2], bout[3]};
#pragma unroll
  for (int k = 0; k < D_H2; ++k) {
    float hv = hr[k];
#pragma unroll
    for (int j = 0; j < D_OUT; ++j) acc[j] += hv * Wout[k * D_OUT + j];
  }
  float m = fmaxf(fmaxf(acc[0], acc[1]), fmaxf(acc[2], acc[3]));
  float e[D_OUT];
  float s = 0.0f;
#pragma unroll
  for (int j = 0; j < D_OUT; ++j) {
    e[j] = __expf(acc[j] - m);
    s += e[j];
  }
  float inv = 1.0f / s;
  float4 o = make_float4(e[0] * inv, e[1] * inv, e[2] * inv, e[3] * inv);
  *(float4*)(out + (size_t)node * D_OUT) = o;
}

// ---------------------------------------------------------------------------
// Launch
// ---------------------------------------------------------------------------
extern "C" void kernel_launch(void* const* d_in, const int* in_sizes, int n_in,
                              void* d_out, int out_size, void* d_ws, size_t ws_size,
                              hipStream_t stream) {
  const float* x    = (const float*)d_in[0];   // 50000*128
  const int*   src  = (const int*)  d_in[1];   // 600000
  const int*   dst  = (const int*)  d_in[2];   // 600000
  const float* w1   = (const float*)d_in[3];   // 128
  const float* Wd1  = (const float*)d_in[4];   // 128*64
  const float* b1   = (const float*)d_in[5];   // 64
  const float* w2   = (const float*)d_in[6];   // 64
  const float* Wd2  = (const float*)d_in[7];   // 64*32
  const float* b2   = (const float*)d_in[8];   // 32
  const float* Wout = (const float*)d_in[9];   // 32*4
  const float* bout = (const float*)d_in[10];  // 4
  float* out = (float*)d_out;                  // 50000*4

  // Workspace carve-up (256B aligned offsets).
  char* ws = (char*)d_ws;
  size_t off = 0;
  auto alloc = [&](size_t bytes) {
    char* p = ws + off;
    off = (off + bytes + 255) & ~(size_t)255;
    return p;
  };
  float* h1   = (float*)alloc((size_t)N_NODES * D_FEAT * 4);  // 25.6 MB
  float* h2   = (float*)alloc((size_t)N_NODES * D_H1   * 4);  // 12.8 MB
  float* h3   = (float*)alloc((size_t)N_NODES * D_H1   * 4);  // 12.8 MB
  float* h4   = (float*)alloc((size_t)N_NODES * D_H2   * 4);  //  6.4 MB
  float* Wd1s = (float*)alloc((size_t)D_FEAT * D_H1 * 4);
  float* Wd2s = (float*)alloc((size_t)D_H1 * D_H2 * 4);
  (void)ws_size; (void)n_in; (void)in_sizes; (void)out_size;

  // 1) Zero the two scatter accumulators.
  zero_kernel<<<2048, 256, 0, stream>>>(h1, N_NODES * D_FEAT);
  zero_kernel<<<2048, 256, 0, stream>>>(h3, N_NODES * D_H1);

  // 2) Fold w1/w2 into the dense weights.
  scale_weights_kernel<<<(D_FEAT * D_H1 + D_H1 * D_H2 + 255) / 256, 256, 0, stream>>>(
      w1, Wd1, w2, Wd2, Wd1s, Wd2s);

  // 3) h1 = scatter_add(x[src] -> dst): one wave per edge, 8 waves/block.
  spmm_scatter128<<<(N_EDGES + 7) / 8, 256, 0, stream>>>(x, src, dst, h1);

  // 4) h2 = h1 @ Wd1s + b1 : M=50000, K=128, N=64 (4 waves/block).
  gemm_wmma_f32<D_FEAT, D_H1><<<N_NODES / 16, 4 * 32, 0, stream>>>(h1, Wd1s, b1, h2);

  // 5) h3 = scatter_add(h2[src] -> dst).
  spmm_scatter64<<<(N_EDGES + 7) / 8, 256, 0, stream>>>(h2, src, dst, h3);

  // 6) h4 = h3 @ Wd2s + b2 : M=50000, K=64, N=32 (2 waves/block).
  gemm_wmma_f32<D_H1, D_H2><<<N_NODES / 16, 2 * 32, 0, stream>>>(h3, Wd2s, b2, h4);

  // 7) logits + softmax.
  out_softmax_kernel<<<(N_NODES + 255) / 256, 256, 0, stream>>>(h4, Wout, bout, out);
}